// ContextCompressor_37168646979630
// MI455X (gfx1250) — compile-verified
//
#include <hip/hip_runtime.h>
#include <hip/hip_bf16.h>

// Problem constants (from reference setup_inputs)
#define BB 8
#define TT 8192
#define DD 1024
#define KK 4096   // k = max(64, T*0.5) = 4096

typedef __attribute__((ext_vector_type(2))) float v2f;
typedef __attribute__((ext_vector_type(8))) float v8f;

// ---------------------------------------------------------------------------
// Kernel 1: scores[b,t] = dot(hidden[b,t,:], query) via V_WMMA_F32_16X16X4_F32
//   A (16x4)  = query chunk replicated on all 16 rows (uniform per lane-half)
//   B (4x16)  = 16 tokens as columns; lane n holds token n's contiguous float2
//   C (16x16) = every row identical; score of token n = c[0] on lane n
// 8 waves / block, 16 tokens / wave -> 128 tokens / block.
// ---------------------------------------------------------------------------
__global__ __launch_bounds__(256)
void cc_scores_wmma(const float* __restrict__ hidden,
                    const float* __restrict__ query,
                    float* __restrict__ scores) {
    __shared__ float q[DD];
    const int tid = threadIdx.x;
    for (int i = tid; i < DD; i += 256) q[i] = query[i];
    __syncthreads();

    const int wave = tid >> 5;
    const int lane = tid & 31;
    const int hi   = lane >> 4;          // 0: K={0,1}, 1: K={2,3} of each step
    const size_t tokBase = ((size_t)blockIdx.x * 8 + wave) * 16;
    const size_t rowOff  = (tokBase + (size_t)(lane & 15)) * DD + (size_t)(hi * 2);

    v8f c = {};
#pragma unroll 8
    for (int kk = 0; kk < DD; kk += 4) {
        // A: query broadcast (uniform within each lane-half) from LDS
        v2f a = *(const v2f*)(q + kk + hi * 2);
        // B: token features, contiguous 8B per lane (8B-aligned: even index)
        v2f b = *(const v2f*)(hidden + rowOff + kk);
        c = __builtin_amdgcn_wmma_f32_16x16x4_f32(
                /*neg_a=*/false, a, /*neg_b=*/false, b,
                /*c_mod=*/(short)0, c, /*reuse_a=*/false, /*reuse_b=*/false);
    }
    if (lane < 16) scores[tokBase + lane] = c[0];
}

// ---------------------------------------------------------------------------
// Kernel 2: exact top-k (k = T/2) per batch row, matching jax.lax.top_k
// tie-break (lowest index first), then ascending-index compaction + mask.
// One 1024-thread block per row; 64KB dynamic LDS bitonic sort of uint64 keys:
//   key = (~orderedBits(score) << 32) | tokenIndex   (ascending sort ==
//   descending score, ascending index on ties; keys are unique)
// ---------------------------------------------------------------------------
__device__ __forceinline__ unsigned long long cc_make_key(float s, int t) {
    unsigned u   = __float_as_uint(s);
    unsigned ord = (u & 0x80000000u) ? ~u : (u | 0x80000000u); // monotone map
    unsigned inv = ~ord;                                       // descending
    return ((unsigned long long)inv << 32) | (unsigned)t;
}

__global__ __launch_bounds__(1024)
void cc_topk_select(const float* __restrict__ scores,
                    int* __restrict__ selIdx,      // [B,K] ascending indices
                    float* __restrict__ mask) {    // [B,T] 0.0/1.0
    extern __shared__ unsigned long long keys[];   // 8192 * 8B = 64KB
    const int b = blockIdx.x;
    const int tid = threadIdx.x;
    const int NTH = 1024;

    for (int t = tid; t < TT; t += NTH)
        keys[t] = cc_make_key(scores[b * TT + t], t);
    __syncthreads();

    // Bitonic sort ascending over 8192 keys
    for (int sz = 2; sz <= TT; sz <<= 1) {
        for (int j = sz >> 1; j > 0; j >>= 1) {
            for (int i = tid; i < TT; i += NTH) {
                int ixj = i ^ j;
                if (ixj > i) {
                    unsigned long long a = keys[i];
                    unsigned long long c = keys[ixj];
                    bool up = ((i & sz) == 0);
                    if (up ? (a > c) : (a < c)) { keys[i] = c; keys[ixj] = a; }
                }
            }
            __syncthreads();
        }
    }

    const unsigned long long kth = keys[KK - 1];   // unique threshold key
    __syncthreads();                               // LDS reuse below

    // Re-scan: thread owns tokens [tid*8, tid*8+8) -> ascending order natural
    int* sums = (int*)keys;
    int sel[8];
    int cnt = 0;
    const int base_t = tid * 8;
#pragma unroll
    for (int i = 0; i < 8; ++i) {
        int t = base_t + i;
        unsigned long long key = cc_make_key(scores[b * TT + t], t);
        sel[i] = (key <= kth) ? 1 : 0;
        cnt += sel[i];
    }
    sums[tid] = cnt;
    __syncthreads();
    // Hillis-Steele inclusive scan over 1024 partials
    for (int off = 1; off < NTH; off <<= 1) {
        int v = sums[tid];
        int add = (tid >= off) ? sums[tid - off] : 0;
        __syncthreads();
        sums[tid] = v + add;
        __syncthreads();
    }
    int pos = sums[tid] - cnt;                     // exclusive prefix
#pragma unroll
    for (int i = 0; i < 8; ++i) {
        int t = base_t + i;
        mask[b * TT + t] = sel[i] ? 1.0f : 0.0f;
        if (sel[i]) { selIdx[b * KK + pos] = t; ++pos; }
    }
}

// ---------------------------------------------------------------------------
// Kernel 3: order-preserving gather, pure b128 bandwidth.
// One block per output row; 256 threads x float4 = 1024 floats.
// ---------------------------------------------------------------------------
__global__ __launch_bounds__(256)
void cc_gather_rows(const float* __restrict__ hidden,
                    const int* __restrict__ selIdx,
                    float* __restrict__ out) {
    const int row = blockIdx.x;                    // 0 .. B*K-1
    const int b   = row >> 12;                     // / KK (4096)
    const int src = selIdx[row];
    const float4* s = (const float4*)(hidden + ((size_t)b * TT + (size_t)src) * DD);
    float4* d = (float4*)(out + (size_t)row * DD);
    d[threadIdx.x] = s[threadIdx.x];
}

extern "C" void kernel_launch(void* const* d_in, const int* in_sizes, int n_in,
                              void* d_out, int out_size, void* d_ws, size_t ws_size,
                              hipStream_t stream) {
    const float* hidden = (const float*)d_in[0];   // (B,T,D) f32
    const float* query  = (const float*)d_in[1];   // (D,)    f32

    float* compressed = (float*)d_out;                         // B*K*D floats
    float* mask = compressed + (size_t)BB * KK * DD;           // B*T floats

    float* scores = (float*)d_ws;                              // B*T floats
    int*   selIdx = (int*)((char*)d_ws + (size_t)BB * TT * sizeof(float)); // B*K ints

    // Pass 1: scores (memory-bound streaming read of hidden, WMMA f32 dot)
    cc_scores_wmma<<<(BB * TT) / 128, 256, 0, stream>>>(hidden, query, scores);

    // Pass 2: exact per-row top-k + ascending compaction + mask (8 blocks)
    cc_topk_select<<<BB, 1024, TT * sizeof(unsigned long long), stream>>>(
        scores, selIdx, mask);

    // Pass 3: gather selected rows (memory-bound)
    cc_gather_rows<<<BB * KK, 256, 0, stream>>>(hidden, selIdx, compressed);
}